// GraphAttention_33131377722144
// MI455X (gfx1250) — compile-verified
//
#include <hip/hip_runtime.h>
#include <math.h>

#define NROWS 4096
#define HIDDEN 512
#define DDIM 64
#define NEG_SLOPE 0.2f
#define KT 64          // K-tile (kernel 1) staged in LDS
#define JT 64          // j-tile (kernel 3) staged in LDS

typedef __attribute__((ext_vector_type(2))) float v2f;
typedef __attribute__((ext_vector_type(4))) float v4f;
typedef __attribute__((ext_vector_type(8))) float v8f;
typedef __attribute__((ext_vector_type(4))) int   v4i;

__device__ __forceinline__ float lrelu(float v) {
    return v > 0.0f ? v : NEG_SLOPE * v;
}

// ---------------------------------------------------------------------------
// Async global->LDS staging (CDNA5 GLOBAL_LOAD_ASYNC_TO_LDS_B128, ASYNCcnt).
// Builtin signature (from hipcc diagnostic): (addrspace(1) v4i*, addrspace(3)
// v4i*, imm offset, imm cpol). Guarded with a register-staging fallback.
// ---------------------------------------------------------------------------
#if __has_builtin(__builtin_amdgcn_global_load_async_to_lds_b128) && \
    __has_builtin(__builtin_amdgcn_s_wait_asynccnt)
#define ASYNC_OK 1
#else
#define ASYNC_OK 0
#endif

typedef __attribute__((address_space(1))) v4i gas_v4i;   // global int4
typedef __attribute__((address_space(3))) v4i las_v4i;   // LDS int4

__device__ __forceinline__ void stage_b128(const float* gsrc, float* lds_dst) {
#if ASYNC_OK
    __builtin_amdgcn_global_load_async_to_lds_b128(
        (gas_v4i*)gsrc, (las_v4i*)lds_dst, 0, 0);
#else
    *(v4f*)lds_dst = *(const v4f*)gsrc;
#endif
}

__device__ __forceinline__ void tile_barrier() {
#if ASYNC_OK
    __builtin_amdgcn_s_wait_asynccnt(0);
#endif
    __syncthreads();
}

// ---------------------------------------------------------------------------
// Kernel 1: xt = x @ W (4096x512 @ 512x64). One block per 16-row tile; wave w
// owns output columns [16w,16w+16); K traversed in 64-wide LDS-staged tiles,
// double-buffered with async copies. f32 WMMA 16x16x4.
// ---------------------------------------------------------------------------
__global__ void __launch_bounds__(128) xt_gemm_kernel(
    const float* __restrict__ x, const float* __restrict__ W,
    float* __restrict__ xt) {
    __shared__ float xs[2][16 * KT];      // x slab: 16 rows x KT     (4 KB each)
    __shared__ float Wt[2][KT * DDIM];    // W slab: KT rows x 64     (16 KB each)

    const int tid   = threadIdx.x;
    const int lane  = tid & 31;
    const int wave  = tid >> 5;
    const int m15   = lane & 15;
    const int khalf = (lane < 16) ? 0 : 2;    // ISA 16x4 f32 A layout
    const int tile  = blockIdx.x;             // 256 tiles of 16 rows
    const int col   = wave * 16 + m15;        // this wave's output column

    auto stage = [&](int buf, int kbase) {
        // x slab: rows tile*16..+15, cols kbase..kbase+KT-1 (2 b128 per thread)
        #pragma unroll
        for (int c = 0; c < 2; ++c) {
            const int chunk = tid + c * 128;          // 0..255
            const int row   = chunk >> 4;
            const int cc    = (chunk & 15) * 4;       // float offset in row
            stage_b128(x + (size_t)(tile * 16 + row) * HIDDEN + kbase + cc,
                       &xs[buf][row * KT + cc]);
        }
        // W slab: KT*DDIM contiguous floats (8 b128 per thread)
        #pragma unroll
        for (int c = 0; c < 8; ++c) {
            const int off = (tid + c * 128) * 4;      // 0..4092
            stage_b128(W + (size_t)kbase * DDIM + off, &Wt[buf][off]);
        }
    };

    v8f acc = {};
    stage(0, 0);
    tile_barrier();

    const int NT = HIDDEN / KT;               // 8 K-tiles
    for (int t = 0; t < NT; ++t) {
        const int cur = t & 1;
        if (t + 1 < NT) stage(cur ^ 1, (t + 1) * KT);   // prefetch next tile
        const float* xsb = xs[cur];
        const float* Wb  = Wt[cur];
        #pragma unroll
        for (int kk = 0; kk < KT; kk += 4) {
            v2f a = *(const v2f*)(xsb + m15 * KT + kk + khalf);   // ds b64
            v2f b = { Wb[(kk + khalf) * DDIM + col],
                      Wb[(kk + khalf + 1) * DDIM + col] };
            acc = __builtin_amdgcn_wmma_f32_16x16x4_f32(
                      false, a, false, b, (short)0, acc, false, false);
        }
        tile_barrier();
    }

    #pragma unroll
    for (int r = 0; r < 8; ++r) {             // C/D: rows r / r+8 by lane half
        const int rowOut = tile * 16 + r + ((lane < 16) ? 0 : 8);
        xt[(size_t)rowOut * DDIM + col] = acc[r];
    }
}

// ---------------------------------------------------------------------------
// Kernel 2a: reset encoded global max (graph-replay deterministic).
// ---------------------------------------------------------------------------
__global__ void init_max_kernel(unsigned* m2u) { *m2u = 0u; }

// Order-preserving f32 <-> u32 encoding for integer atomicMax.
__device__ __forceinline__ unsigned enc_f32(float f) {
    unsigned b = __float_as_uint(f);
    return (b & 0x80000000u) ? ~b : (b | 0x80000000u);
}
__device__ __forceinline__ float dec_f32(unsigned u) {
    unsigned b = (u & 0x80000000u) ? (u & 0x7FFFFFFFu) : ~u;
    return __uint_as_float(b);
}

// ---------------------------------------------------------------------------
// Kernel 2: s1 = xt@a1, s2 = xt@a2, M2 = max(s2). One row per thread.
// ---------------------------------------------------------------------------
__global__ void __launch_bounds__(256) score_kernel(
    const float* __restrict__ xt, const float* __restrict__ a,
    float* __restrict__ s1, float* __restrict__ s2, unsigned* m2u) {
    __shared__ float smax[256];
    const int row = blockIdx.x * 256 + threadIdx.x;
    const float* r = xt + (size_t)row * DDIM;
    float acc1 = 0.0f, acc2 = 0.0f;
    #pragma unroll
    for (int k = 0; k < DDIM; ++k) {
        float v = r[k];
        acc1 = fmaf(v, a[k], acc1);
        acc2 = fmaf(v, a[DDIM + k], acc2);
    }
    s1[row] = acc1;
    s2[row] = acc2;

    smax[threadIdx.x] = acc2;
    __syncthreads();
    for (int off = 128; off > 0; off >>= 1) {
        if (threadIdx.x < off)
            smax[threadIdx.x] = fmaxf(smax[threadIdx.x], smax[threadIdx.x + off]);
        __syncthreads();
    }
    if (threadIdx.x == 0) atomicMax(m2u, enc_f32(smax[0]));
}

// ---------------------------------------------------------------------------
// Kernel 3: fused softmax-GEMM. m_i = lrelu(s1_i + max(s2)) is the EXACT row
// max (leaky_relu is monotone), so no online rescaling. One block per 16-row
// tile; wave w owns output columns [16w,16w+16); the xt j-tile (64x64) and the
// s2 slab are async-staged into LDS, double-buffered, shared by all 4 waves.
// Attention weights (A operand, ISA 16x4 layout) built in registers.
// ---------------------------------------------------------------------------
__global__ void __launch_bounds__(128) attn_kernel(
    const float* __restrict__ xt, const float* __restrict__ s1,
    const float* __restrict__ s2, const unsigned* __restrict__ m2u,
    float* __restrict__ out) {
    __shared__ float jt[2][JT * DDIM];    // xt j-slab (16 KB each)
    __shared__ float s2t[2][JT];          // s2 slab   (256 B each)

    const int tid   = threadIdx.x;
    const int lane  = tid & 31;
    const int wave  = tid >> 5;
    const int m15   = lane & 15;
    const int khalf = (lane < 16) ? 0 : 2;
    const int tile  = blockIdx.x;             // 256 tiles of 16 rows
    const int col   = wave * 16 + m15;

    const float M2   = dec_f32(*m2u);
    const float s1v  = s1[tile * 16 + m15];   // this lane's row score
    const float mrow = lrelu(s1v + M2);       // exact row max

    auto stage = [&](int buf, int jbase) {
        #pragma unroll
        for (int c = 0; c < 8; ++c) {         // 16 KB / 128 threads / 16 B
            const int off = (tid + c * 128) * 4;
            stage_b128(xt + (size_t)jbase * DDIM + off, &jt[buf][off]);
        }
        if (tid < 16)                         // 64 floats of s2
            stage_b128(s2 + jbase + tid * 4, &s2t[buf][tid * 4]);
    };

    v8f acc = {};
    float dsum = 0.0f;
    stage(0, 0);
    tile_barrier();

    const int NT = NROWS / JT;                // 64 j-tiles
    for (int t = 0; t < NT; ++t) {
        const int cur = t & 1;
        if (t + 1 < NT) stage(cur ^ 1, (t + 1) * JT);   // prefetch next tile
        const float* jb = jt[cur];
        const float* sb = s2t[cur];
        #pragma unroll
        for (int jj = 0; jj < JT; jj += 4) {
            const int jl = jj + khalf;
            v2f sv = *(const v2f*)(sb + jl);            // ds b64
            float w0 = __expf(lrelu(s1v + sv.x) - mrow);
            float w1 = __expf(lrelu(s1v + sv.y) - mrow);
            v2f aw = { w0, w1 };
            dsum += w0 + w1;
            v2f b = { jb[jl * DDIM + col], jb[(jl + 1) * DDIM + col] };
            acc = __builtin_amdgcn_wmma_f32_16x16x4_f32(
                      false, aw, false, b, (short)0, acc, false, false);
        }
        tile_barrier();
    }

    // Row m15 lives in lanes m15 and m15+16: fold the two K-halves.
    dsum += __shfl_xor(dsum, 16, 32);         // lanes 0..15 hold denom[row]

    #pragma unroll
    for (int r = 0; r < 8; ++r) {
        const int rit = r + ((lane < 16) ? 0 : 8);
        const float denom = __shfl(dsum, rit, 32);
        out[(size_t)(tile * 16 + rit) * DDIM + col] = acc[r] / denom;
    }
}

// ---------------------------------------------------------------------------
// Launch: x(4096x512) f32, W(512x64) f32, a(128) f32 -> out(4096x64) f32.
// Workspace: xt (1 MB) | s1 (16 KB) | s2 (16 KB) | m2u (4 B).
// ---------------------------------------------------------------------------
extern "C" void kernel_launch(void* const* d_in, const int* in_sizes, int n_in,
                              void* d_out, int out_size, void* d_ws, size_t ws_size,
                              hipStream_t stream) {
    const float* x = (const float*)d_in[0];
    const float* W = (const float*)d_in[1];
    const float* a = (const float*)d_in[2];
    float* out = (float*)d_out;

    float*    xt  = (float*)d_ws;
    float*    s1  = xt + (size_t)NROWS * DDIM;
    float*    s2  = s1 + NROWS;
    unsigned* m2u = (unsigned*)(s2 + NROWS);

    xt_gemm_kernel<<<NROWS / 16, 128, 0, stream>>>(x, W, xt);
    init_max_kernel<<<1, 1, 0, stream>>>(m2u);
    score_kernel<<<NROWS / 256, 256, 0, stream>>>(xt, a, s1, s2, m2u);
    attn_kernel<<<NROWS / 16, 128, 0, stream>>>(xt, s1, s2, m2u, out);
}